// TokenMapper_59940563583540
// MI455X (gfx1250) — compile-verified
//
#include <hip/hip_runtime.h>

typedef float v2f __attribute__((ext_vector_type(2)));
typedef float v4f __attribute__((ext_vector_type(4)));
typedef float v8f __attribute__((ext_vector_type(8)));

#define VAE    768
#define OUTD   1024
#define PARTS  16
#define ROWSPP 257                 // NUM_K + 1
#define ROWS   (ROWSPP * PARTS)    // 4112
#define MB     2                   // M-tiles per wave
#define NB     4                   // N-tiles per wave
#define MBLK   ((ROWS / 16 + MB - 1) / MB)   // 129 (last block half-valid)
#define NBLK   (OUTD / 16 / NB)              // 16

// ---------------------------------------------------------------------------
// Tiled transpose through LDS (32x33 padding -> bank-conflict free).
// out[C][R] = in[R][C]^T.  R, C multiples of 32.
// ---------------------------------------------------------------------------
__global__ __launch_bounds__(256) void tm_transpose(
    const float* __restrict__ in, float* __restrict__ out, int R, int C) {
  __shared__ float tile[32][33];
  const int x  = blockIdx.x * 32 + threadIdx.x;   // col of in
  const int y0 = blockIdx.y * 32 + threadIdx.y;   // row of in
#pragma unroll
  for (int j = 0; j < 32; j += 8)
    tile[threadIdx.y + j][threadIdx.x] = in[(size_t)(y0 + j) * C + x];
  __syncthreads();
  const int xo = blockIdx.y * 32 + threadIdx.x;   // col of out (= row of in)
  const int yo = blockIdx.x * 32 + threadIdx.y;   // row of out (= col of in)
#pragma unroll
  for (int j = 0; j < 32; j += 8)
    out[(size_t)(yo + j) * R + xo] = tile[threadIdx.x][threadIdx.y + j];
}

// ---------------------------------------------------------------------------
// Register-blocked f32 WMMA GEMM core: one wave owns MB x NB 16x16 tiles.
// A:[ROWSA,K] row-major, BT:[OUTD,K] (pre-transposed W) -> both fragments are
// single contiguous b64 loads per lane:
//   A frag : lane = m + 16*(k/2), float2 = A[m][k+2h .. k+2h+1]
//   B frag : lane = n + 16*(k/2), float2 = BT[n][k+2h .. k+2h+1]
// C/D: vgpr r -> M = r + 8*(lane>>4), N = lane&15
// Per K-step: MB+NB = 6 b64 loads feed MB*NB = 8 WMMAs.
// ---------------------------------------------------------------------------
template <int K, bool RELU, bool ADD_PE>
__device__ __forceinline__ void gemm_core(
    const float* __restrict__ A, const float* __restrict__ BT,
    const float* __restrict__ bias, const float* __restrict__ pe,
    float* __restrict__ C) {
  const int wave  = (blockIdx.x << 3) + (threadIdx.x >> 5);
  const int lane  = threadIdx.x & 31;
  const int mb    = wave / NBLK;
  const int nb    = wave - mb * NBLK;
  const int l15   = lane & 15;
  const int khalf = lane >> 4;

  // Clamp second M-tile's row into range (results discarded by store guard).
  const int r0 = mb * (16 * MB) + l15;
  const int r1 = min(r0 + 16, ROWS - 1);
  const float* __restrict__ a0 = A + (size_t)r0 * K + 2 * khalf;
  const float* __restrict__ a1 = A + (size_t)r1 * K + 2 * khalf;
  const float* __restrict__ bp =
      BT + (size_t)(nb * (16 * NB) + l15) * K + 2 * khalf;

  v8f acc[MB][NB] = {};
#pragma unroll 2
  for (int k = 0; k < K; k += 4) {
    v2f af[MB], bf[NB];
    af[0] = *(const v2f*)(a0 + k);
    af[1] = *(const v2f*)(a1 + k);
#pragma unroll
    for (int j = 0; j < NB; ++j)
      bf[j] = *(const v2f*)(bp + k + (size_t)j * 16 * K);  // const ioffset
#pragma unroll
    for (int i = 0; i < MB; ++i)
#pragma unroll
      for (int j = 0; j < NB; ++j)
        acc[i][j] = __builtin_amdgcn_wmma_f32_16x16x4_f32(
            false, af[i], false, bf[j], (short)0, acc[i][j], false, false);
  }

#pragma unroll
  for (int i = 0; i < MB; ++i) {
    if (mb * (16 * MB) + i * 16 >= ROWS) break;   // wave-uniform guard
#pragma unroll
    for (int j = 0; j < NB; ++j) {
      const int n  = nb * (16 * NB) + j * 16 + l15;
      const float bn = bias[n];
#pragma unroll
      for (int r = 0; r < 8; ++r) {
        const int row = mb * (16 * MB) + i * 16 + r + 8 * khalf;
        float v = acc[i][j][r] + bn;
        if (RELU) v = v > 0.0f ? v : 0.0f;
        if (ADD_PE) v += pe[(size_t)(row / ROWSPP) * OUTD + n];
        C[(size_t)row * OUTD + n] = v;
      }
    }
  }
}

// GEMM1: table_h = relu(emb[4112,768] @ W1 + b1)     (W1 pre-transposed)
__global__ __launch_bounds__(256) void tm_gemm1_relu(
    const float* __restrict__ A, const float* __restrict__ BT,
    const float* __restrict__ bias, float* __restrict__ C) {
  gemm_core<VAE, true, false>(A, BT, bias, nullptr, C);
}

// GEMM2: table_out = table_h @ W2 + b2 + pe[row/257] (W2 pre-transposed)
__global__ __launch_bounds__(256) void tm_gemm2_pe(
    const float* __restrict__ A, const float* __restrict__ BT,
    const float* __restrict__ bias, const float* __restrict__ pe,
    float* __restrict__ C) {
  gemm_core<OUTD, false, true>(A, BT, bias, pe, C);
}

// ---------------------------------------------------------------------------
// Gather: out[bp,:] = table_out[hashes[bp] + 257*(bp&15), :]
// Table (16.8 MB) stays L2-resident; output written non-temporally so the
// 268 MB stream does not evict the table.
// ---------------------------------------------------------------------------
__global__ __launch_bounds__(256) void tm_gather(
    const int* __restrict__ hashes, const float* __restrict__ table,
    float* __restrict__ out) {
  const int bp = blockIdx.x;                    // 0 .. B*PARTS-1
  const int p  = bp & (PARTS - 1);
  const int h  = hashes[bp];                    // uniform -> scalar load
  const v4f v =
      ((const v4f*)table)[(size_t)(p * ROWSPP + h) * (OUTD / 4) + threadIdx.x];
  __builtin_nontemporal_store(
      v, (v4f*)out + (size_t)bp * (OUTD / 4) + threadIdx.x);
}

extern "C" void kernel_launch(void* const* d_in, const int* in_sizes, int n_in,
                              void* d_out, int out_size, void* d_ws, size_t ws_size,
                              hipStream_t stream) {
  const int*   hashes = (const int*)d_in[0];
  const float* emb    = (const float*)d_in[1];
  const float* W1     = (const float*)d_in[2];
  const float* b1     = (const float*)d_in[3];
  const float* W2     = (const float*)d_in[4];
  const float* b2     = (const float*)d_in[5];
  const float* pe     = (const float*)d_in[6];
  float*       out    = (float*)d_out;

  // Workspace layout (~24.2 MB): table_out | W1T | W2T.
  float* table_out = (float*)d_ws;                      // 4112*1024
  float* W1T       = table_out + (size_t)ROWS * OUTD;   // 1024*768
  float* W2T       = W1T + (size_t)OUTD * VAE;          // 1024*1024
  float* table_h   = out;   // staged in d_out, fully overwritten by gather

  dim3 tb(32, 8);
  tm_transpose<<<dim3(OUTD / 32, VAE / 32),  tb, 0, stream>>>(W1, W1T, VAE,  OUTD);
  tm_transpose<<<dim3(OUTD / 32, OUTD / 32), tb, 0, stream>>>(W2, W2T, OUTD, OUTD);

  const int waves  = MBLK * NBLK;   // 129*16 = 2064
  const int blocks = waves / 8;     // 258 blocks x 8 waves, exact

  tm_gemm1_relu<<<blocks, 256, 0, stream>>>(emb, W1T, b1, table_h);
  tm_gemm2_pe  <<<blocks, 256, 0, stream>>>(table_h, W2T, b2, pe, table_out);
  tm_gather    <<<4096 * PARTS, 256, 0, stream>>>(hashes, table_out, out);
}